// GNN_41601053229788
// MI455X (gfx1250) — compile-verified
//
#include <hip/hip_runtime.h>
#include <math.h>

#define NVAR 50000
#define NCON 50000
#define DD   128
#define DIN  32
#define DEG  16
#define KTOT (2 * DD + DIN)   // 288

typedef __attribute__((ext_vector_type(2))) float v2f;
typedef __attribute__((ext_vector_type(8))) float v8f;

__device__ __forceinline__ v8f wmma4(v2f a, v2f b, v8f c) {
  return __builtin_amdgcn_wmma_f32_16x16x4_f32(
      /*neg_a=*/false, a, /*neg_b=*/false, b,
      /*c_mod=*/(short)0, c, /*reuse_a=*/false, /*reuse_b=*/false);
}

// Wt[n*K + k] = W[k*N + n]  (one-time weight transpose so B loads are b64)
__global__ void transpose_w_kernel(const float* __restrict__ W, float* __restrict__ Wt,
                                   int K, int N) {
  int i = blockIdx.x * blockDim.x + threadIdx.x;
  if (i >= K * N) return;
  int k = i / N, n = i % N;
  Wt[(size_t)n * K + k] = W[i];
}

// ---- single-tile K-segment, transposed B (init GEMM) -----------------------
// Abase: row base (contiguous K). WtCol: Wt + nCol*KW (per-lane column base).
template<int KW>
__device__ __forceinline__ v8f wmma_segT(const float* __restrict__ Abase,
                                         const float* __restrict__ WtCol,
                                         int hv, v8f acc) {
#pragma unroll
  for (int k = 0; k < KW; k += 4) {
    v2f a = *(const v2f*)(Abase + k + 2 * hv);   // A: lanes0-15 K=k..k+1, lanes16-31 K=k+2..k+3
    v2f b = *(const v2f*)(WtCol + k + 2 * hv);   // B: same K split -> contiguous in Wt
    acc = wmma4(a, b, acc);
  }
  return acc;
}

// ---- 2 row-tiles x 4 col-tiles K-segment, transposed B (layer GEMMs) -------
// 6 b64 loads per 8 WMMAs.
template<int KW>
__device__ __forceinline__ void wmma_seg2x4T(const float* __restrict__ A0,
                                             const float* __restrict__ A1,
                                             const float* __restrict__ w0,
                                             const float* __restrict__ w1,
                                             const float* __restrict__ w2,
                                             const float* __restrict__ w3,
                                             int hv, v8f acc[2][4]) {
#pragma unroll
  for (int k = 0; k < KW; k += 4) {
    v2f a0 = *(const v2f*)(A0 + k + 2 * hv);
    v2f a1 = *(const v2f*)(A1 + k + 2 * hv);
    v2f b0 = *(const v2f*)(w0 + k + 2 * hv);
    v2f b1 = *(const v2f*)(w1 + k + 2 * hv);
    v2f b2 = *(const v2f*)(w2 + k + 2 * hv);
    v2f b3 = *(const v2f*)(w3 + k + 2 * hv);
    acc[0][0] = wmma4(a0, b0, acc[0][0]);  acc[1][0] = wmma4(a1, b0, acc[1][0]);
    acc[0][1] = wmma4(a0, b1, acc[0][1]);  acc[1][1] = wmma4(a1, b1, acc[1][1]);
    acc[0][2] = wmma4(a0, b2, acc[0][2]);  acc[1][2] = wmma4(a1, b2, acc[1][2]);
    acc[0][3] = wmma4(a0, b3, acc[0][3]);  acc[1][3] = wmma4(a1, b3, acc[1][3]);
  }
}

// lv = x[:NVAR] @ Wv + bv ; lc = x[NVAR:] @ Wc + bc   (K = 32, 1 tile/wave)
// Wtv/Wtc are transposed weights: (128, 32)
__global__ void init_gemm_kernel(const float* __restrict__ x,
                                 const float* __restrict__ Wtv, const float* __restrict__ bv,
                                 const float* __restrict__ Wtc, const float* __restrict__ bc,
                                 float* __restrict__ lv, float* __restrict__ lc) {
  int wave = threadIdx.x >> 5;
  int tile = blockIdx.x * (blockDim.x >> 5) + wave;
  const int totalTiles = ((NVAR + NCON) / 16) * (DD / 16);
  if (tile >= totalTiles) return;
  int colTile = tile & 7;
  int rowTile = tile >> 3;
  int lane = threadIdx.x & 31;
  int hv   = lane >> 4;
  int l16  = lane & 15;
  int row  = rowTile * 16 + l16;
  int nCol = colTile * 16 + l16;
  bool isV = rowTile < (NVAR / 16);   // NVAR multiple of 16: tiles never straddle
  const float* Wt = isV ? Wtv : Wtc;
  const float* bb = isV ? bv : bc;
  v8f acc = {};
  acc = wmma_segT<DIN>(x + (size_t)row * DIN, Wt + (size_t)nCol * DIN, hv, acc);
  float badd = bb[nCol];
  float* outp = isV ? lv : lc;
  int mbase = rowTile * 16 - (isV ? 0 : NVAR);
#pragma unroll
  for (int r = 0; r < 8; ++r)
    outp[(size_t)(mbase + r + 8 * hv) * DD + nCol] = acc[r] + badd;
}

// out = concat([Aagg, Acur, Afeat], dim=1) @ W + bias  (K = 288)
// Wt: transposed weights (128, 288). Each wave: 32x64 output block (2x4 tiles).
__global__ void layer_gemm_kernel(const float* __restrict__ Aagg,
                                  const float* __restrict__ Acur,
                                  const float* __restrict__ Afeat,
                                  const float* __restrict__ Wt,
                                  const float* __restrict__ bias,
                                  float* __restrict__ out, int M) {
  int wave = threadIdx.x >> 5;
  int task = blockIdx.x * (blockDim.x >> 5) + wave;
  int rowTiles = M / 16;                 // 3125
  int rowBlks  = (rowTiles + 1) / 2;     // 1563 (last one partial)
  int totalTasks = rowBlks * 2;          // 2 col-blocks of 64 columns
  if (task >= totalTasks) return;
  int colBlk = task & 1;
  int rowBlk = task >> 1;
  int lane = threadIdx.x & 31;
  int hv   = lane >> 4;
  int l16  = lane & 15;
  int rt0  = rowBlk * 2;
  int rt1  = rt0 + 1;
  bool v1  = rt1 < rowTiles;             // wave-uniform validity of 2nd row tile
  int rtS  = v1 ? rt1 : rt0;             // safe row tile for loads
  int row0 = rt0 * 16 + l16;
  int row1 = rtS * 16 + l16;
  int nCol = colBlk * 64 + l16;          // + j*16, j in 0..3

  // Per-column-tile weight base pointers (immediate offsets inside the K loop)
  const float* w0 = Wt + (size_t)(nCol + 0 * 16) * KTOT;
  const float* w1 = Wt + (size_t)(nCol + 1 * 16) * KTOT;
  const float* w2 = Wt + (size_t)(nCol + 2 * 16) * KTOT;
  const float* w3 = Wt + (size_t)(nCol + 3 * 16) * KTOT;

  v8f acc[2][4] = {};
  wmma_seg2x4T<DD >(Aagg  + (size_t)row0 * DD,  Aagg  + (size_t)row1 * DD,
                    w0, w1, w2, w3, hv, acc);
  wmma_seg2x4T<DD >(Acur  + (size_t)row0 * DD,  Acur  + (size_t)row1 * DD,
                    w0 + DD, w1 + DD, w2 + DD, w3 + DD, hv, acc);
  wmma_seg2x4T<DIN>(Afeat + (size_t)row0 * DIN, Afeat + (size_t)row1 * DIN,
                    w0 + 2 * DD, w1 + 2 * DD, w2 + 2 * DD, w3 + 2 * DD, hv, acc);

#pragma unroll
  for (int i = 0; i < 2; ++i) {
    if (i == 1 && !v1) break;            // wave-uniform skip of dangling tile
    int mbase = (rt0 + i) * 16;
#pragma unroll
    for (int j = 0; j < 4; ++j) {
      float badd = bias[nCol + j * 16];
#pragma unroll
      for (int r = 0; r < 8; ++r)
        out[(size_t)(mbase + r + 8 * hv) * DD + nCol + j * 16] = acc[i][j][r] + badd;
    }
  }
}

// dst[row] = sum_{j<16} src[idx[row][j]]   (one block per row, 128 threads)
__global__ void agg_kernel(const float* __restrict__ src, const int* __restrict__ idx,
                           float* __restrict__ dst) {
  __shared__ int sidx[DEG];
  int row = blockIdx.x;
  if (threadIdx.x < DEG) sidx[threadIdx.x] = idx[(size_t)row * DEG + threadIdx.x];
  __syncthreads();
  int col = threadIdx.x;
  float s = 0.f;
#pragma unroll
  for (int j = 0; j < DEG; ++j) s += src[(size_t)sidx[j] * DD + col];
  dst[(size_t)row * DD + col] = s;
}

// aggv[col] = sum_rows lv[row][col]   (one block per column, deterministic tree)
__global__ void colsum_kernel(const float* __restrict__ lv, float* __restrict__ aggv) {
  __shared__ float red[256];
  int col = blockIdx.x;
  float s = 0.f;
  for (int r = threadIdx.x; r < NVAR; r += 256) s += lv[(size_t)r * DD + col];
  red[threadIdx.x] = s;
  __syncthreads();
  for (int off = 128; off > 0; off >>= 1) {
    if (threadIdx.x < off) red[threadIdx.x] += red[threadIdx.x + off];
    __syncthreads();
  }
  if (threadIdx.x == 0) aggv[col] = red[0];
}

// Q[i] = dot(aggv, Wq[:128]) + dot(lv[i], Wq[128:]) + bq ; inf-mask on trunc(x[i,1])!=0
__global__ void q_kernel(const float* __restrict__ lv, const float* __restrict__ aggv,
                         const float* __restrict__ Wq, const float* __restrict__ bq,
                         const float* __restrict__ x, float* __restrict__ out) {
  int i = blockIdx.x * blockDim.x + threadIdx.x;
  if (i >= NVAR) return;
  float s = bq[0];
#pragma unroll 8
  for (int k = 0; k < DD; ++k) s += aggv[k] * Wq[k];
#pragma unroll 8
  for (int k = 0; k < DD; ++k) s += lv[(size_t)i * DD + k] * Wq[DD + k];
  float f = x[(size_t)i * DIN + 1];
  out[i] = (truncf(f) != 0.0f) ? __builtin_inff() : s;
}

extern "C" void kernel_launch(void* const* d_in, const int* in_sizes, int n_in,
                              void* d_out, int out_size, void* d_ws, size_t ws_size,
                              hipStream_t stream) {
  const float* x    = (const float*)d_in[0];
  const int*   vci  = (const int*)d_in[1];   // var_constr_index (NVAR, 16)
  const int*   cvi  = (const int*)d_in[2];   // constr_var_index (NCON, 16)
  const float* Wiv  = (const float*)d_in[3];
  const float* biv  = (const float*)d_in[4];
  const float* Wic  = (const float*)d_in[5];
  const float* bic  = (const float*)d_in[6];
  const float* Wvar = (const float*)d_in[7];
  const float* bvar = (const float*)d_in[8];
  const float* Wcon = (const float*)d_in[9];
  const float* bcon = (const float*)d_in[10];
  const float* Wq   = (const float*)d_in[11];
  const float* bq   = (const float*)d_in[12];
  float* out = (float*)d_out;

  // Workspace: 5 x (50000*128) floats + 128 + 4 transposed weight buffers (~128.3 MB)
  const size_t NVD = (size_t)NVAR * DD;
  float* lv0    = (float*)d_ws;
  float* lc0    = lv0 + NVD;
  float* lv1    = lc0 + NVD;
  float* lc1    = lv1 + NVD;
  float* aggbuf = lc1 + NVD;
  float* aggvec = aggbuf + NVD;          // 128
  float* WtIv   = aggvec + DD;           // (128, 32)
  float* WtIc   = WtIv + DIN * DD;       // (128, 32)
  float* WtCon  = WtIc + DIN * DD;       // (128, 288)
  float* WtVar  = WtCon + (size_t)KTOT * DD;

  // One-time weight transposes (tiny; stream-ordered, deterministic)
  transpose_w_kernel<<<(DIN * DD + 255) / 256, 256, 0, stream>>>(Wiv, WtIv, DIN, DD);
  transpose_w_kernel<<<(DIN * DD + 255) / 256, 256, 0, stream>>>(Wic, WtIc, DIN, DD);
  transpose_w_kernel<<<(KTOT * DD + 255) / 256, 256, 0, stream>>>(Wcon, WtCon, KTOT, DD);
  transpose_w_kernel<<<(KTOT * DD + 255) / 256, 256, 0, stream>>>(Wvar, WtVar, KTOT, DD);

  // Init layer: 6250 row-tiles x 8 col-tiles, 8 waves (tiles) per 256-thread block
  {
    int tiles  = ((NVAR + NCON) / 16) * (DD / 16);
    int blocks = (tiles + 7) / 8;
    init_gemm_kernel<<<blocks, 256, 0, stream>>>(x, WtIv, biv, WtIc, bic, lv0, lc0);
  }

  float* lvc = lv0; float* lcc = lc0; float* lvn = lv1; float* lcn = lc1;
  const float* cfeat = x + (size_t)NVAR * DIN;
  const int rowTiles = NCON / 16;                // 3125 (same for NVAR)
  const int tasks    = ((rowTiles + 1) / 2) * 2; // 2x4-tile blocks per GEMM
  const int lblocks  = (tasks + 7) / 8;

  for (int layer = 0; layer < 3; ++layer) {
    // c_agg = gather-sum of lv over constr_var_index; new_lc = [c_agg, lc, cfeat] @ Wcon + bcon
    agg_kernel<<<NCON, 128, 0, stream>>>(lvc, cvi, aggbuf);
    layer_gemm_kernel<<<lblocks, 256, 0, stream>>>(aggbuf, lcc, cfeat, WtCon, bcon, lcn, NCON);
    // v_agg = gather-sum of OLD lc over var_constr_index; new_lv = [v_agg, lv, vfeat] @ Wvar + bvar
    agg_kernel<<<NVAR, 128, 0, stream>>>(lcc, vci, aggbuf);
    layer_gemm_kernel<<<lblocks, 256, 0, stream>>>(aggbuf, lvc, x, WtVar, bvar, lvn, NVAR);
    float* t;
    t = lvc; lvc = lvn; lvn = t;
    t = lcc; lcc = lcn; lcn = t;
  }

  colsum_kernel<<<DD, 256, 0, stream>>>(lvc, aggvec);
  q_kernel<<<(NVAR + 255) / 256, 256, 0, stream>>>(lvc, aggvec, Wq, bq, x, out);
}